// P4DConv_68436008895044
// MI455X (gfx1250) — compile-verified
//
#include <hip/hip_runtime.h>
#include <hip/hip_bf16.h>
#include <stdint.h>

typedef __attribute__((ext_vector_type(2))) float v2f;
typedef __attribute__((ext_vector_type(8))) float v8f;
typedef __attribute__((ext_vector_type(4))) unsigned int u32x4;
typedef __attribute__((ext_vector_type(8))) int i32x8;
typedef __attribute__((ext_vector_type(4))) int i32x4;

#define B_   4
#define T_   32
#define N_   4096
#define F_   16
#define M_   128
#define K_   32
#define C0_  64
#define C1_  128
#define R2_  (0.15f * 0.15f)

// ---------------------------------------------------------------------------
// Kernel 1: furthest point sampling. One block per group g = b*F + f.
// center frame (original coords) = 2*f. Writes anchor xyz into d_out[0..G*M*3).
// Matches jnp.argmax first-index tie-break (lower index wins on equal dist).
// ---------------------------------------------------------------------------
__global__ __launch_bounds__(256) void fps_kernel(const float* __restrict__ xyzs,
                                                  float* __restrict__ out_xyz) {
    const int g   = blockIdx.x;          // 0..63
    const int b   = g / F_;
    const int f   = g % F_;
    const int tc  = 2 * f;               // center frame in original coords
    const float* pts = xyzs + (size_t)(b * T_ + tc) * N_ * 3;
    const int tid = threadIdx.x;         // 0..255
    constexpr int PPT = N_ / 256;        // 16 points per thread

    __shared__ float sdist[256];
    __shared__ int   sidx[256];
    __shared__ float ssel[3];

    // Cache this thread's points in registers (strided -> coalesced loads;
    // p = j*256 + tid is monotone in j so ascending-j scan keeps lowest index).
    float px[PPT], py[PPT], pz[PPT], dist[PPT];
#pragma unroll
    for (int j = 0; j < PPT; ++j) {
        int p = j * 256 + tid;
        px[j] = pts[p * 3 + 0];
        py[j] = pts[p * 3 + 1];
        pz[j] = pts[p * 3 + 2];
        dist[j] = 1e10f;
    }

    if (tid == 0) {
        float x = pts[0], y = pts[1], z = pts[2];
        out_xyz[(g * M_ + 0) * 3 + 0] = x;
        out_xyz[(g * M_ + 0) * 3 + 1] = y;
        out_xyz[(g * M_ + 0) * 3 + 2] = z;
        ssel[0] = x; ssel[1] = y; ssel[2] = z;
    }
    __syncthreads();
    float lx = ssel[0], ly = ssel[1], lz = ssel[2];

    for (int it = 1; it < M_; ++it) {
        float best = -1.0f;
        int   bidx = 0x7fffffff;
#pragma unroll
        for (int j = 0; j < PPT; ++j) {
            float dx = px[j] - lx, dy = py[j] - ly, dz = pz[j] - lz;
            float d2 = dx * dx + dy * dy + dz * dz;
            dist[j] = fminf(dist[j], d2);
            if (dist[j] > best) { best = dist[j]; bidx = j * 256 + tid; }
        }
        sdist[tid] = best;
        sidx[tid]  = bidx;
        __syncthreads();
        for (int s = 128; s > 0; s >>= 1) {
            if (tid < s) {
                float od = sdist[tid + s];
                int   oi = sidx[tid + s];
                if (od > sdist[tid] || (od == sdist[tid] && oi < sidx[tid])) {
                    sdist[tid] = od; sidx[tid] = oi;
                }
            }
            __syncthreads();
        }
        if (tid == 0) {
            int sel = sidx[0];
            float x = pts[sel * 3 + 0], y = pts[sel * 3 + 1], z = pts[sel * 3 + 2];
            out_xyz[(g * M_ + it) * 3 + 0] = x;
            out_xyz[(g * M_ + it) * 3 + 1] = y;
            out_xyz[(g * M_ + it) * 3 + 2] = z;
            ssel[0] = x; ssel[1] = y; ssel[2] = z;
        }
        __syncthreads();
        lx = ssel[0]; ly = ssel[1]; lz = ssel[2];
    }
}

// ---------------------------------------------------------------------------
// Kernel 2: fused ball-query + gather + MLP(4->64->128, ReLU) + max_k + sum_dt.
// One wave32 per (g, m); 2 waves per 64-thread block. Wm (32 KB) is DMA'd into
// LDS once per block by the Tensor Data Mover, overlapping the ball-query scan.
// WMMA f32 16x16x4 layout assumptions (CDNA5 ISA 7.12.2):
//   A(m,k): lane = (m&15) + 16*(k>=2), vgpr = k&1
//   B(k,n): lane = (n&15) + 16*(k>=2), vgpr = k&1
//   C(m,n): vgpr = m&7,  lane = (n&15) + 16*((m>>3)&1)   (per 16x16 tile)
// ---------------------------------------------------------------------------
__global__ __launch_bounds__(64) void p4d_fused_kernel(const float* __restrict__ xyzs,
                                                       const float* __restrict__ Wd,
                                                       const float* __restrict__ Wm,
                                                       const float* __restrict__ anchors,
                                                       float* __restrict__ out_feat) {
    const int lane  = threadIdx.x & 31;
    const int warp  = threadIdx.x >> 5;                 // 0..1
    const int wave  = blockIdx.x * 2 + warp;            // 0..8191
    const int g     = wave >> 7;                        // 0..63
    const int m     = wave & 127;                       // anchor index
    const int b     = g >> 4;
    const int f     = g & 15;

    __shared__ float sWm[C1_ * C0_];       // 128x64 weights, TDM-loaded (32 KB)
    __shared__ float sD[2][K_ * 4];        // 32x4 displacement matrices
    __shared__ float sH[2][K_ * C0_];      // 32x64 hidden activations
    __shared__ int   sIdx[2][K_];          // ball-query neighbor indices

    // ---- kick off async TDM copy of Wm -> LDS (wave 0 only; EXEC ignored) ----
    if (warp == 0) {
        uint32_t lds_base = (uint32_t)(uintptr_t)&sWm[0];
        uint64_t gaddr    = (uint64_t)(uintptr_t)Wm;
        u32x4 g0;
        g0[0] = 1u;                                   // count=1, user descriptor
        g0[1] = lds_base;                             // lds_addr
        g0[2] = (uint32_t)gaddr;                      // global_addr[31:0]
        g0[3] = (uint32_t)((gaddr >> 32) & 0x1FFFFFFu) | 0x80000000u; // addr[56:32] | type=2
        i32x8 g1;
        g1[0] = 0x00020000;                           // wg_mask=0, data_size=2 (4B)
        g1[1] = (int)(8192u << 16);                   // tensor_dim0[15:0]=8192 in [63:48]
        g1[2] = (int)(1u << 16);                      // tensor_dim0 hi=0; tensor_dim1=1
        g1[3] = (int)(8192u << 16);                   // tensor_dim1 hi=0; tile_dim0=8192
        g1[4] = 1;                                    // tile_dim1=1, tile_dim2=0
        g1[5] = 8192;                                 // tensor_dim0_stride lo
        g1[6] = 0;                                    // stride hi / dim1_stride lo
        g1[7] = 0;
        i32x4 gz = {0, 0, 0, 0};
#if defined(__clang_major__) && __clang_major__ >= 23
        i32x8 gz8 = {0, 0, 0, 0, 0, 0, 0, 0};
        __builtin_amdgcn_tensor_load_to_lds(g0, g1, gz, gz, gz8, 0);
#else
        __builtin_amdgcn_tensor_load_to_lds(g0, g1, gz, gz, 0);
#endif
    }

    const float ax = anchors[(g * M_ + m) * 3 + 0];
    const float ay = anchors[(g * M_ + m) * 3 + 1];
    const float az = anchors[(g * M_ + m) * 3 + 2];

    const int  half = (lane >= 16) ? 2 : 0;   // K-pair selector for A/B frags
    const int  lcol = lane & 15;

    // Preload Wd^T B-fragments: 4 N-tiles of 16 output channels each.
    float bw0[4], bw1[4];
#pragma unroll
    for (int n = 0; n < 4; ++n) {
        int o = n * 16 + lcol;                 // output channel 0..63
        bw0[n] = Wd[o * 4 + half + 0];
        bw1[n] = Wd[o * 4 + half + 1];
    }

    float featsum[8];
#pragma unroll
    for (int n = 0; n < 8; ++n) featsum[n] = 0.0f;

    for (int dt = -1; dt <= 1; ++dt) {
        int tn = 2 * f + dt;
        if (tn < 0) tn = 0;                    // temporal edge padding (top edge never exceeds 31)
        const float* np = xyzs + (size_t)(b * T_ + tn) * N_ * 3;

        // ---- ball query: first K hits in ascending global index order ----
        sIdx[warp][lane] = 0;                  // default for the "no hits" case
        __syncthreads();

        int running = 0;                       // uniform across the wave
        for (int i = 0; i < N_ / 32 && running < K_; ++i) {
            int p = i * 32 + lane;             // (i, lane) lexicographic == ascending p
            float dx = np[p * 3 + 0] - ax;
            float dy = np[p * 3 + 1] - ay;
            float dz = np[p * 3 + 2] - az;
            float d2 = dx * dx + dy * dy + dz * dz;
            bool hit = d2 < R2_;
            unsigned mask = (unsigned)__ballot(hit);
            if (hit) {
                int pos = running + __popc(mask & ((1u << lane) - 1u));
                if (pos < K_) sIdx[warp][pos] = p;
            }
            running += __popc(mask);
        }
        __syncthreads();

        int vcnt  = running < K_ ? running : K_;
        int first = sIdx[warp][0];             // stays 0 if no hits (matches reference)
        int myidx = (lane < vcnt) ? sIdx[warp][lane] : first;

        // ---- gather + displacement: lane l owns neighbor-slot row l of D ----
        float gx = np[myidx * 3 + 0];
        float gy = np[myidx * 3 + 1];
        float gz = np[myidx * 3 + 2];
        sD[warp][lane * 4 + 0] = gx - ax;
        sD[warp][lane * 4 + 1] = gy - ay;
        sD[warp][lane * 4 + 2] = gz - az;
        sD[warp][lane * 4 + 3] = (float)dt;
        __syncthreads();

        // ---- layer 1: H(32x64) = relu(D(32x4) @ Wd^T(4x64)), one WMMA per tile ----
        v2f a0, a1;
        a0[0] = sD[warp][(0  + lcol) * 4 + half + 0];
        a0[1] = sD[warp][(0  + lcol) * 4 + half + 1];
        a1[0] = sD[warp][(16 + lcol) * 4 + half + 0];
        a1[1] = sD[warp][(16 + lcol) * 4 + half + 1];

#pragma unroll
        for (int n = 0; n < 4; ++n) {
            v2f bw; bw[0] = bw0[n]; bw[1] = bw1[n];
            v8f cz = {};
            v8f h0 = __builtin_amdgcn_wmma_f32_16x16x4_f32(false, a0, false, bw,
                                                           (short)0, cz, false, false);
            v8f h1 = __builtin_amdgcn_wmma_f32_16x16x4_f32(false, a1, false, bw,
                                                           (short)0, cz, false, false);
#pragma unroll
            for (int v = 0; v < 8; ++v) {
                int mr = v + ((lane >= 16) ? 8 : 0);        // row within 16-row tile
                int oc = n * 16 + lcol;
                sH[warp][(mr +  0) * C0_ + oc] = fmaxf(h0[v], 0.0f);
                sH[warp][(mr + 16) * C0_ + oc] = fmaxf(h1[v], 0.0f);
            }
        }
        // Make sure the TDM transfer of Wm has landed before anyone reads sWm.
        if (warp == 0) __builtin_amdgcn_s_wait_tensorcnt(0);
        __syncthreads();

        // ---- layer 2: H(32x64) @ Wm^T(64x128), relu, max over the 32 rows ----
#pragma unroll 1
        for (int n = 0; n < 8; ++n) {
            v8f acc0 = {};
            v8f acc1 = {};
            const int o = n * 16 + lcol;       // output channel 0..127
#pragma unroll 4
            for (int ks = 0; ks < 16; ++ks) {
                int k = ks * 4 + half;
                v2f fa0, fa1, bb;
                fa0[0] = sH[warp][( 0 + lcol) * C0_ + k + 0];
                fa0[1] = sH[warp][( 0 + lcol) * C0_ + k + 1];
                fa1[0] = sH[warp][(16 + lcol) * C0_ + k + 0];
                fa1[1] = sH[warp][(16 + lcol) * C0_ + k + 1];
                bb[0]  = sWm[o * C0_ + k + 0];
                bb[1]  = sWm[o * C0_ + k + 1];
                acc0 = __builtin_amdgcn_wmma_f32_16x16x4_f32(false, fa0, false, bb,
                                                             (short)0, acc0, false, false);
                acc1 = __builtin_amdgcn_wmma_f32_16x16x4_f32(false, fa1, false, bb,
                                                             (short)0, acc1, false, false);
            }
            // column max over 32 rows: 8 VGPRs per lane cover rows {0..7} or {8..15}
            // of each 16-row tile; shfl_xor(16) merges the two lane halves.
            float cm = acc0[0];
#pragma unroll
            for (int v = 1; v < 8; ++v) cm = fmaxf(cm, acc0[v]);
#pragma unroll
            for (int v = 0; v < 8; ++v) cm = fmaxf(cm, acc1[v]);
            cm = fmaxf(cm, __shfl_xor(cm, 16, 32));
            cm = fmaxf(cm, 0.0f);              // relu commutes with max
            featsum[n] += cm;
        }
        __syncthreads();
    }

    // new_features layout: (B, F, C1, M) -> flat ((g*C1 + o)*M + m)
    if (lane < 16) {
#pragma unroll
        for (int n = 0; n < 8; ++n) {
            int o = n * 16 + lane;
            out_feat[((size_t)g * C1_ + o) * M_ + m] = featsum[n];
        }
    }
}

// ---------------------------------------------------------------------------
extern "C" void kernel_launch(void* const* d_in, const int* in_sizes, int n_in,
                              void* d_out, int out_size, void* d_ws, size_t ws_size,
                              hipStream_t stream) {
    (void)in_sizes; (void)n_in; (void)out_size; (void)d_ws; (void)ws_size;

    const float* xyzs = (const float*)d_in[0];   // (4,32,4096,3)
    const float* Wd   = (const float*)d_in[1];   // (64,4)
    const float* Wm   = (const float*)d_in[2];   // (128,64)

    float* out      = (float*)d_out;
    float* out_xyz  = out;                               // (B,F,M,3) = 24576 floats
    float* out_feat = out + (size_t)B_ * F_ * M_ * 3;    // (B,F,C1,M) = 262144 floats

    // 64 groups, one block each
    fps_kernel<<<B_ * F_, 256, 0, stream>>>(xyzs, out_xyz);

    // 64*128 = 8192 waves, 2 waves (64 threads) per block
    p4d_fused_kernel<<<(B_ * F_ * M_) / 2, 64, 0, stream>>>(xyzs, Wd, Wm, out_xyz, out_feat);
}